// Repro_71545565217337
// MI455X (gfx1250) — compile-verified
//
#include <hip/hip_runtime.h>
#include <hip/hip_bf16.h>
#include <math.h>

typedef __attribute__((ext_vector_type(2))) float v2f;
typedef __attribute__((ext_vector_type(8))) float v8f;

namespace {
constexpr int Nb = 64, Cc = 3, Hh = 345, Ww = 456;
constexpr int TH = (Hh + 15) / 16;   // 22 tile rows
constexpr int TW = (Ww + 15) / 16;   // 29 tile cols
constexpr int TILES = Nb * TH * TW;  // 40832 -> exactly 5104 blocks * 8 waves
constexpr int HW = Hh * Ww;
// torch affine_grid (align_corners=False) linspace: xs[i] = i*step + start
// (both symmetric branches of the reference reduce to this in exact math;
// using the same step/start constants keeps FP semantics close).
constexpr float STEPX = 2.0f / (float)Ww;
constexpr float STARTX = -(1.0f - 1.0f / (float)Ww);
constexpr float STEPY = 2.0f / (float)Hh;
constexpr float STARTY = -(1.0f - 1.0f / (float)Hh);
}

// One wave32 per 16x16 spatial tile. Coordinates for the whole tile are
// produced by two V_WMMA_F32_16X16X4_F32 ops (ix and iy); each lane then
// owns 8 pixels (same w, h = v + 8*(lane>=16)) and performs the bilinear
// gather for all 3 channels. Output uses non-temporal stores (write-once
// stream) so L2 stays dedicated to the gathered input image.
__global__ __launch_bounds__(256) void grid_sample_wmma_kernel(
    const float* __restrict__ x, const float* __restrict__ theta,
    float* __restrict__ out) {
  const int lane = threadIdx.x & 31;
  const int wid = blockIdx.x * (blockDim.x >> 5) + (threadIdx.x >> 5);
  if (wid >= TILES) return;  // wave-uniform; grid sized exactly anyway

  const int n = wid / (TH * TW);
  const int t = wid - n * (TH * TW);
  const int th = t / TW;
  const int tw = t - th * TW;
  const int h0 = th << 4, w0 = tw << 4;

  // theta row 0 -> gx, row 1 -> gy; fold the (align_corners=False)
  // unnormalization ix = gx*W/2 + (W-1)/2 into the affine coefficients.
  const float* tp = theta + n * 6;
  const float sx0 = tp[0] * (0.5f * Ww), sx1 = tp[1] * (0.5f * Ww);
  const float cx = tp[2] * (0.5f * Ww) + 0.5f * (Ww - 1);
  const float sy0 = tp[3] * (0.5f * Hh), sy1 = tp[4] * (0.5f * Hh);
  const float cy = tp[5] * (0.5f * Hh) + 0.5f * (Hh - 1);

  const bool lo = lane < 16;
  const int lm = lane & 15;
  const float xsw = fmaf((float)(w0 + lm), STEPX, STARTX);
  const float ysh = fmaf((float)(h0 + lm), STEPY, STARTY);

  // 32-bit A 16x4 layout: lanes 0-15 hold K=0 (V0), K=1 (V1); lanes 16-31
  // hold K=2,K=3 -> we zero those rows so only K0/K1 contribute.
  //   A[m,0] = ys[h0+m], A[m,1] = 1
  //   B[0,n] = s1 coeff, B[1,n] = s0*xs[w0+n] + c
  // => D[m,n] = ys[m]*s1 + s0*xs[n] + c   (the unnormalized sample coord)
  v2f A, Bx, By;
  A[0] = lo ? ysh : 0.0f;
  A[1] = lo ? 1.0f : 0.0f;
  Bx[0] = lo ? sx1 : 0.0f;
  Bx[1] = lo ? fmaf(sx0, xsw, cx) : 0.0f;
  By[0] = lo ? sy1 : 0.0f;
  By[1] = lo ? fmaf(sy0, xsw, cy) : 0.0f;
  v8f zc = {0.0f, 0.0f, 0.0f, 0.0f, 0.0f, 0.0f, 0.0f, 0.0f};

  v8f IX = __builtin_amdgcn_wmma_f32_16x16x4_f32(false, A, false, Bx,
                                                 (short)0, zc, false, false);
  v8f IY = __builtin_amdgcn_wmma_f32_16x16x4_f32(false, A, false, By,
                                                 (short)0, zc, false, false);

  // D layout: lane L, VGPR v  <->  pixel (h = v + 8*(L>=16), w = L%16)
  const int w = w0 + lm;
  const int hb = h0 + (lo ? 0 : 8);
  const float* xb = x + (size_t)n * Cc * HW;
  float* ob = out + (size_t)n * Cc * HW;
  const bool wok = w < Ww;

#pragma unroll
  for (int v = 0; v < 8; ++v) {
    const int h = hb + v;
    if (!(wok && h < Hh)) continue;  // tile padding (divergence is post-WMMA)
    const float ix = IX[v], iy = IY[v];
    const float fx = floorf(ix), fy = floorf(iy);
    const int x0i = (int)fx, y0i = (int)fy;
    const float ax = ix - fx, ay = iy - fy;
    const float bx_ = 1.0f - ax, by_ = 1.0f - ay;
    // zeros padding: clamp index into range, zero the corner weight
    const bool vx0 = (x0i >= 0) & (x0i < Ww);
    const bool vx1 = (x0i >= -1) & (x0i < Ww - 1);
    const bool vy0 = (y0i >= 0) & (y0i < Hh);
    const bool vy1 = (y0i >= -1) & (y0i < Hh - 1);
    const int xi0 = vx0 ? x0i : 0, xi1 = vx1 ? x0i + 1 : 0;
    const int yi0 = vy0 ? y0i : 0, yi1 = vy1 ? y0i + 1 : 0;
    const float m00 = (vx0 & vy0) ? bx_ * by_ : 0.0f;
    const float m10 = (vx1 & vy0) ? ax * by_ : 0.0f;
    const float m01 = (vx0 & vy1) ? bx_ * ay : 0.0f;
    const float m11 = (vx1 & vy1) ? ax * ay : 0.0f;
    const int i00 = yi0 * Ww + xi0, i10 = yi0 * Ww + xi1;
    const int i01 = yi1 * Ww + xi0, i11 = yi1 * Ww + xi1;
    const int ho = h * Ww + w;
#pragma unroll
    for (int c = 0; c < Cc; ++c) {
      const float* p = xb + c * HW;
      float r = m00 * p[i00];
      r = fmaf(m10, p[i10], r);
      r = fmaf(m01, p[i01], r);
      r = fmaf(m11, p[i11], r);
      __builtin_nontemporal_store(r, ob + c * HW + ho);  // NT: keep L2 for x
    }
  }
}

extern "C" void kernel_launch(void* const* d_in, const int* in_sizes, int n_in,
                              void* d_out, int out_size, void* d_ws,
                              size_t ws_size, hipStream_t stream) {
  (void)in_sizes; (void)n_in; (void)out_size; (void)d_ws; (void)ws_size;
  const float* x = (const float*)d_in[0];      // [64,3,345,456] f32
  const float* theta = (const float*)d_in[1];  // [64,2,3] f32
  float* out = (float*)d_out;                  // [64,3,345,456] f32
  const int wavesPerBlock = 256 / 32;          // 8 waves per block
  const int blocks = (TILES + wavesPerBlock - 1) / wavesPerBlock;  // 5104
  grid_sample_wmma_kernel<<<blocks, 256, 0, stream>>>(x, theta, out);
}